// BuildPatches_65309272703438
// MI455X (gfx1250) — compile-verified
//
#include <hip/hip_runtime.h>
#include <cstdint>
#include <cstddef>

typedef __attribute__((ext_vector_type(2))) float v2f;
typedef __attribute__((ext_vector_type(8))) float v8f;

// Problem dims (from setup_inputs)
constexpr int B = 8;
constexpr int N = 16384;
constexpr int R = 2048;
constexpr int P = 32;

// Flat output offsets (floats / int32 slots), concatenated in return order
constexpr size_t OFF_PATCHES = 0;                              // B*R*P*3
constexpr size_t OFF_IDX     = (size_t)B * R * P * 3;          // B*R*P*2 (int32)
constexpr size_t OFF_DIST    = OFF_IDX + (size_t)B * R * P * 2;// B*R*P
constexpr size_t OFF_MAT     = OFF_DIST + (size_t)B * R * P;   // B*R*N

// ---------------- Kernel 1: distance matrix via V_WMMA_F32_16X16X4_F32 -----
// dist[m][n] = ||r_m||^2 - 2 r_m.p_n + ||p_n||^2
// A (16x4):  row m = (-2rx, -2ry, -2rz, 1)      -> lanes 0-15: K0/K1, lanes 16-31: K2/K3
// B (4x16):  col n = ( px,  py,  pz, ||p||^2 )  -> same K split across half-waves
// C (16x16): c[j] = ||r_{base+j+8*half}||^2     -> per ISA C/D layout
constexpr int TILES_PER_WAVE = 64;                 // 1024 columns per wave
constexpr int NCHUNK = N / (16 * TILES_PER_WAVE);  // 16
constexpr int RT = R / 16;                         // 128

__global__ __launch_bounds__(128) void dist_mat_wmma(
    const float* __restrict__ points, const float* __restrict__ roots,
    float* __restrict__ mat) {
  const int wave = (blockIdx.x << 2) | (threadIdx.x >> 5);
  const int lane = threadIdx.x & 31;
  const int chunk = wave % NCHUNK;
  const int rt = (wave / NCHUNK) % RT;
  const int b = wave / (NCHUNK * RT);
  const int half = lane >> 4;
  const int m16 = lane & 15;

  const int rbase = rt * 16;
  const float* rp = roots + ((size_t)b * R + rbase + m16) * 3;
  const float rx = rp[0], ry = rp[1], rz = rp[2];

  v2f a;
  a.x = half ? (-2.0f * rz) : (-2.0f * rx);  // K=2 : K=0
  a.y = half ? 1.0f : (-2.0f * ry);          // K=3 : K=1

  v8f c;
#pragma unroll
  for (int j = 0; j < 8; ++j) {
    const float* rj = roots + ((size_t)b * R + rbase + j + half * 8) * 3;
    const float x = rj[0], y = rj[1], z = rj[2];
    c[j] = x * x + y * y + z * z;  // ||r||^2 for the row this slot maps to
  }

  float* out = mat + (size_t)b * R * N;
  const int ncol0 = chunk * (16 * TILES_PER_WAVE);
  for (int t = 0; t < TILES_PER_WAVE; ++t) {
    const int n0 = ncol0 + t * 16;
    const float* pp = points + ((size_t)b * N + n0 + m16) * 3;
    const float px = pp[0], py = pp[1], pz = pp[2];
    const float r1 = px * px + py * py + pz * pz;
    v2f bb;
    bb.x = half ? pz : px;  // K=2 : K=0
    bb.y = half ? r1 : py;  // K=3 : K=1
    v8f d = __builtin_amdgcn_wmma_f32_16x16x4_f32(
        false, a, false, bb, (short)0, c, false, false);
#pragma unroll
    for (int j = 0; j < 8; ++j) {
      const int row = rbase + j + half * 8;
      out[(size_t)row * N + n0 + m16] = d[j];
    }
  }
}

// ---------------- Kernel 2: fused recompute + top-32 per root --------------
// One wave32 per (b,r) row. Lane-private top-32 lists live in LDS with a
// [entry*32 + lane] layout (bank-conflict-free, no barriers needed since each
// lane touches only its own slots). Selection emits ascending distances.
__global__ __launch_bounds__(128) void topk_patches(
    const float* __restrict__ points, const float* __restrict__ roots,
    float* __restrict__ out_patches, int* __restrict__ out_idx,
    float* __restrict__ out_dist) {
  __shared__ float s_val[4][32 * 32];
  __shared__ int s_idx[4][32 * 32];
  const int w = threadIdx.x >> 5;
  const int lane = threadIdx.x & 31;
  const int row = (blockIdx.x << 2) | w;
  const int b = row / R;

  const float FINF = __builtin_inff();
  const float* rp = roots + (size_t)row * 3;
  const float rx = rp[0], ry = rp[1], rz = rp[2];
  const float r0 = rx * rx + ry * ry + rz * rz;

  float* sv = s_val[w];
  int* si = s_idx[w];

#pragma unroll
  for (int e = 0; e < 32; ++e) {
    sv[e * 32 + lane] = FINF;
    si[e * 32 + lane] = 0;
  }

  float vmax = FINF;
  int pmax = 0;
  const float* pbase = points + (size_t)b * N * 3;
  for (int i = 0; i < N / 32; ++i) {
    const int n = i * 32 + lane;
    const float* pp = pbase + (size_t)n * 3;
    const float px = pp[0], py = pp[1], pz = pp[2];
    const float dist =
        r0 - 2.0f * (rx * px + ry * py + rz * pz) + (px * px + py * py + pz * pz);
    if (dist < vmax) {
      sv[pmax * 32 + lane] = dist;
      si[pmax * 32 + lane] = n;
      vmax = -FINF;
#pragma unroll
      for (int e = 0; e < 32; ++e) {
        const float v = sv[e * 32 + lane];
        if (v > vmax) { vmax = v; pmax = e; }
      }
    }
  }

  // per-lane current minimum
  float lmin = FINF;
  int lpos = 0;
#pragma unroll
  for (int e = 0; e < 32; ++e) {
    const float v = sv[e * 32 + lane];
    if (v < lmin) { lmin = v; lpos = e; }
  }

  for (int t = 0; t < P; ++t) {
    float v = lmin;
    int who = lane;
#pragma unroll
    for (int off = 16; off > 0; off >>= 1) {
      const float ov = __shfl_xor(v, off, 32);
      const int owho = __shfl_xor(who, off, 32);
      if (ov < v || (ov == v && owho < who)) { v = ov; who = owho; }
    }
    if (lane == who) {
      const int n = si[lpos * 32 + lane];
      const size_t o = (size_t)row * P + t;
      out_dist[o] = v;
      out_idx[o * 2 + 0] = b;
      out_idx[o * 2 + 1] = n;
      const float* pp = pbase + (size_t)n * 3;
      out_patches[o * 3 + 0] = pp[0] - rx;
      out_patches[o * 3 + 1] = pp[1] - ry;
      out_patches[o * 3 + 2] = pp[2] - rz;
      sv[lpos * 32 + lane] = FINF;
      lmin = FINF;
      lpos = 0;
#pragma unroll
      for (int e = 0; e < 32; ++e) {
        const float vv = sv[e * 32 + lane];
        if (vv < lmin) { lmin = vv; lpos = e; }
      }
    }
  }
}

extern "C" void kernel_launch(void* const* d_in, const int* in_sizes, int n_in,
                              void* d_out, int out_size, void* d_ws,
                              size_t ws_size, hipStream_t stream) {
  (void)in_sizes; (void)n_in; (void)out_size; (void)d_ws; (void)ws_size;
  const float* points = (const float*)d_in[0];
  const float* roots = (const float*)d_in[1];
  float* out = (float*)d_out;

  // Kernel 1: B*RT*NCHUNK waves = 16384, 4 waves/block
  dist_mat_wmma<<<(B * RT * NCHUNK) / 4, 128, 0, stream>>>(points, roots,
                                                           out + OFF_MAT);
  // Kernel 2: B*R rows = 16384, 4 rows/block
  topk_patches<<<(B * R) / 4, 128, 0, stream>>>(
      points, roots, out + OFF_PATCHES, (int*)(out + OFF_IDX), out + OFF_DIST);
}